// LightGCNBaseline_38792144617774
// MI455X (gfx1250) — compile-verified
//
#include <hip/hip_runtime.h>
#include <math.h>

#define N_NODES   100000
#define N_EDGES   1600000
#define DIM       128
#define N_CLASSES 2
#define N_LAYERS  3

typedef __attribute__((ext_vector_type(2))) float v2f;
typedef __attribute__((ext_vector_type(8))) float v8f;

// ---------------------------------------------------------------------------
// softmax over the (tiny) alpha vector: one thread, n <= 8
// ---------------------------------------------------------------------------
__global__ void lgcn_softmax_small(const float* __restrict__ alpha,
                                   float* __restrict__ out, int n) {
  if (blockIdx.x == 0 && threadIdx.x == 0) {
    float m = alpha[0];
    for (int i = 1; i < n; ++i) m = fmaxf(m, alpha[i]);
    float s = 0.0f;
    for (int i = 0; i < n; ++i) s += expf(alpha[i] - m);
    float inv = 1.0f / s;
    for (int i = 0; i < n; ++i) out[i] = expf(alpha[i] - m) * inv;
  }
}

// ---------------------------------------------------------------------------
// Pack the wave-invariant padded B fragments for the WMMA projection.
// bfrag[kt][lane][0..1] = {Wpad[k0][n], Wpad[k0+1][n]}, k0 = 4*kt + 2*(lane>>4),
// n = lane&15, Wpad is W (128x2) zero-padded to 128x16. 32*32*2 floats = 8KB.
// ---------------------------------------------------------------------------
__global__ void lgcn_pack_bfrag(const float* __restrict__ W,
                                float* __restrict__ bfrag) {
  int t = blockIdx.x * blockDim.x + threadIdx.x;  // 0..1023
  if (t >= 32 * 32) return;
  int kt   = t >> 5;
  int lane = t & 31;
  int half = lane >> 4;
  int m    = lane & 15;
  int k0   = kt * 4 + half * 2;
  float bx = (m < N_CLASSES) ? W[(k0 + 0) * N_CLASSES + m] : 0.0f;
  float by = (m < N_CLASSES) ? W[(k0 + 1) * N_CLASSES + m] : 0.0f;
  bfrag[(size_t)t * 2 + 0] = bx;
  bfrag[(size_t)t * 2 + 1] = by;
}

// ---------------------------------------------------------------------------
// x = embedding[node_indices]; cur = x; acc = a[0] * x   (one float4 / thread)
// ---------------------------------------------------------------------------
__global__ void lgcn_gather_init(const int* __restrict__ idx,
                                 const float* __restrict__ emb,
                                 const float* __restrict__ aw,
                                 float* __restrict__ cur,
                                 float* __restrict__ acc) {
  const int VPN = DIM / 4;  // 32 float4 per node
  int t = blockIdx.x * blockDim.x + threadIdx.x;
  if (t >= N_NODES * VPN) return;
  int node = t >> 5;
  int j    = t & 31;
  int src  = idx[node];
  float4 x = reinterpret_cast<const float4*>(emb)[(size_t)src * VPN + j];
  reinterpret_cast<float4*>(cur)[t] = x;
  float a0 = aw[0];
  float4 y; y.x = a0 * x.x; y.y = a0 * x.y; y.z = a0 * x.z; y.w = a0 * x.w;
  reinterpret_cast<float4*>(acc)[t] = y;
}

// ---------------------------------------------------------------------------
// zero fill (float4 granularity)
// ---------------------------------------------------------------------------
__global__ void lgcn_fill_zero(float4* __restrict__ p, int n4) {
  int t = blockIdx.x * blockDim.x + threadIdx.x;
  if (t < n4) { float4 z; z.x = z.y = z.z = z.w = 0.0f; p[t] = z; }
}

// ---------------------------------------------------------------------------
// SpMM scatter: one wave per edge; lane handles 4 consecutive feature dims.
// next[row] += val * cur[col]  via native f32 atomics (L2-resident data).
// ---------------------------------------------------------------------------
__global__ void lgcn_spmm_scatter(const int* __restrict__ rows,
                                  const int* __restrict__ cols,
                                  const float* __restrict__ vals,
                                  const float* __restrict__ cur,
                                  float* __restrict__ nxt) {
  int wavesPerBlock = blockDim.x >> 5;
  int e = blockIdx.x * wavesPerBlock + (threadIdx.x >> 5);
  if (e >= N_EDGES) return;
  int lane = threadIdx.x & 31;
  int r = rows[e];
  int c = cols[e];
  float v = vals[e];
  float4 x = reinterpret_cast<const float4*>(cur)[(size_t)c * (DIM / 4) + lane];
  float* dst = nxt + (size_t)r * DIM + lane * 4;
  __hip_atomic_fetch_add(dst + 0, v * x.x, __ATOMIC_RELAXED, __HIP_MEMORY_SCOPE_AGENT);
  __hip_atomic_fetch_add(dst + 1, v * x.y, __ATOMIC_RELAXED, __HIP_MEMORY_SCOPE_AGENT);
  __hip_atomic_fetch_add(dst + 2, v * x.z, __ATOMIC_RELAXED, __HIP_MEMORY_SCOPE_AGENT);
  __hip_atomic_fetch_add(dst + 3, v * x.w, __ATOMIC_RELAXED, __HIP_MEMORY_SCOPE_AGENT);
}

// ---------------------------------------------------------------------------
// acc += a[ai] * next   (float4 granularity)
// ---------------------------------------------------------------------------
__global__ void lgcn_axpy_acc(const float* __restrict__ aw, int ai,
                              const float* __restrict__ nxt,
                              float* __restrict__ acc) {
  int t = blockIdx.x * blockDim.x + threadIdx.x;
  if (t >= N_NODES * (DIM / 4)) return;
  float a = aw[ai];
  float4 x = reinterpret_cast<const float4*>(nxt)[t];
  float4 y = reinterpret_cast<float4*>(acc)[t];
  y.x += a * x.x; y.y += a * x.y; y.z += a * x.z; y.w += a * x.w;
  reinterpret_cast<float4*>(acc)[t] = y;
}

// ---------------------------------------------------------------------------
// Final projection out[N,2] = acc[N,128] @ W[128,2] + b  via WMMA f32 16x16x4.
// One wave computes a 16-row tile; B fragments come pre-packed from bfrag
// (wave-invariant), so the inner loop is 2 unconditional b64 loads + 1 WMMA
// with no EXEC divergence.
// A frag (16x4 f32): lanes 0-15 hold K=0,1 ; lanes 16-31 hold K=2,3 (row M=lane&15).
// C/D (16x16 f32, 8 VGPRs): VGPR r -> M=r (lanes 0-15), M=r+8 (lanes 16-31), N=lane&15.
// ---------------------------------------------------------------------------
__global__ void lgcn_proj_wmma(const float* __restrict__ acc,
                               const float* __restrict__ bfrag,
                               const float* __restrict__ bias,
                               float* __restrict__ out) {
  int wave = blockIdx.x * (blockDim.x >> 5) + (threadIdx.x >> 5);
  const int nWaves = N_NODES / 16;  // 6250, exact
  if (wave >= nWaves) return;
  int lane = threadIdx.x & 31;
  int half = lane >> 4;     // 0: K pair {0,1}; 1: K pair {2,3}
  int m    = lane & 15;     // A: row within tile ; B/C/D: column n
  int row  = wave * 16 + m;

  const v2f* aptr = reinterpret_cast<const v2f*>(acc + (size_t)row * DIM + half * 2);
  const v2f* bptr = reinterpret_cast<const v2f*>(bfrag) + lane;

  v8f c = {};
#pragma unroll 8
  for (int kt = 0; kt < DIM / 4; ++kt) {
    v2f a = aptr[kt * 2];    // acc[row][4*kt + 2*half .. +1]
    v2f b = bptr[kt * 32];   // bfrag[kt][lane][0..1]
    c = __builtin_amdgcn_wmma_f32_16x16x4_f32(false, a, false, b,
                                              (short)0, c, false, false);
  }

  if (m < N_CLASSES) {
    float bb = bias[m];
    int baseRow = wave * 16 + half * 8;
#pragma unroll
    for (int r = 0; r < 8; ++r) {
      out[(size_t)(baseRow + r) * N_CLASSES + m] = c[r] + bb;
    }
  }
}

// ---------------------------------------------------------------------------
// launch
// ---------------------------------------------------------------------------
extern "C" void kernel_launch(void* const* d_in, const int* in_sizes, int n_in,
                              void* d_out, int out_size, void* d_ws, size_t ws_size,
                              hipStream_t stream) {
  (void)n_in; (void)out_size; (void)ws_size;
  const int*   node_idx = (const int*)d_in[0];
  const int*   adj_row  = (const int*)d_in[1];
  const int*   adj_col  = (const int*)d_in[2];
  const float* adj_val  = (const float*)d_in[3];
  const float* emb      = (const float*)d_in[4];
  const float* alpha    = (const float*)d_in[5];
  const float* W        = (const float*)d_in[6];
  const float* bias     = (const float*)d_in[7];
  float*       out      = (float*)d_out;

  char* ws = (char*)d_ws;
  float* aw    = (float*)ws;             // softmax(alpha), 4 floats
  float* bfrag = (float*)(ws + 256);     // packed B fragments, 8 KB
  size_t base  = 256 + 32 * 32 * 2 * sizeof(float);   // 8448, 256-aligned
  size_t ndBytes = (size_t)N_NODES * DIM * sizeof(float);  // 51.2 MB
  float* acc = (float*)(ws + base);
  float* cur = (float*)(ws + base + ndBytes);
  float* nxt = (float*)(ws + base + 2 * ndBytes);

  int nAlpha = in_sizes[5];  // NUM_LAYERS + 1
  lgcn_softmax_small<<<1, 32, 0, stream>>>(alpha, aw, nAlpha);
  lgcn_pack_bfrag<<<1, 1024, 0, stream>>>(W, bfrag);

  const int n4 = N_NODES * (DIM / 4);
  lgcn_gather_init<<<(n4 + 255) / 256, 256, 0, stream>>>(node_idx, emb, aw, cur, acc);

  for (int layer = 0; layer < N_LAYERS; ++layer) {
    lgcn_fill_zero<<<(n4 + 255) / 256, 256, 0, stream>>>((float4*)nxt, n4);
    const int edgesPerBlock = 8;  // 256 threads = 8 waves, one wave per edge
    lgcn_spmm_scatter<<<(N_EDGES + edgesPerBlock - 1) / edgesPerBlock, 256, 0, stream>>>(
        adj_row, adj_col, adj_val, cur, nxt);
    lgcn_axpy_acc<<<(n4 + 255) / 256, 256, 0, stream>>>(aw, layer + 1, nxt, acc);
    float* tmp = cur; cur = nxt; nxt = tmp;
  }

  const int nWaves = N_NODES / 16;   // 6250
  const int wavesPerBlock = 8;
  lgcn_proj_wmma<<<(nWaves + wavesPerBlock - 1) / wavesPerBlock, 256, 0, stream>>>(
      acc, bfrag, bias, out);
}